// DefenceWrapper_25091198943700
// MI455X (gfx1250) — compile-verified
//
#include <hip/hip_runtime.h>
#include <stdint.h>

#define NUM_CLASSES 1000
#define BLOCK 256
#define PAD 1024           // 256 threads * 4 elements

typedef float v2f __attribute__((ext_vector_type(2)));
typedef float v8f __attribute__((ext_vector_type(8)));

#define NEGBIG (-1.0e30f)
#define OUT_LO (-16.118095651f)   // log(EPS/C) = log(1e-7)
#define OUT_HI (-9.990499e-05f)   // log(1 - EPS + EPS/C) = log(0.9999001)

__device__ __forceinline__ float waveSum(float v) {
#pragma unroll
    for (int m = 16; m > 0; m >>= 1) v += __shfl_xor(v, m, 32);
    return v;
}
__device__ __forceinline__ float waveMax(float v) {
#pragma unroll
    for (int m = 16; m > 0; m >>= 1) v = fmaxf(v, __shfl_xor(v, m, 32));
    return v;
}

__device__ __forceinline__ float blockSum(float v, float* sRed) {
    v = waveSum(v);
    const int w = threadIdx.x >> 5, l = threadIdx.x & 31;
    if (l == 0) sRed[w] = v;
    __syncthreads();
    if (w == 0) {
        float t = (l < 8) ? sRed[l] : 0.0f;
#pragma unroll
        for (int m = 4; m > 0; m >>= 1) t += __shfl_xor(t, m, 32);
        if (l == 0) sRed[0] = t;
    }
    __syncthreads();
    float r = sRed[0];
    __syncthreads();
    return r;
}

__device__ __forceinline__ float blockMax(float v, float* sRed) {
    v = waveMax(v);
    const int w = threadIdx.x >> 5, l = threadIdx.x & 31;
    if (l == 0) sRed[w] = v;
    __syncthreads();
    if (w == 0) {
        float t = (l < 8) ? sRed[l] : NEGBIG;
#pragma unroll
        for (int m = 4; m > 0; m >>= 1) t = fmaxf(t, __shfl_xor(t, m, 32));
        if (l == 0) sRed[0] = t;
    }
    __syncthreads();
    float r = sRed[0];
    __syncthreads();
    return r;
}

__global__ void __launch_bounds__(BLOCK)
sanitize_sample_kernel(const float* __restrict__ logits,
                       const float* __restrict__ noise,
                       const float* __restrict__ u,
                       float* __restrict__ out) {
    __shared__ float sLog[PAD];
    __shared__ float sNoise[PAD];
    __shared__ float sPart[BLOCK];
    __shared__ float sScan[BLOCK];
    __shared__ float sRow[16];
    __shared__ float sRed[8];

    const int row = blockIdx.x;
    const int t   = threadIdx.x;
    const int j0  = t * 4;

    // ---- Stage row into LDS via gfx1250 async copy (ASYNCcnt path) ----
    const float* gl = logits + (size_t)row * NUM_CLASSES;
    const float* gn = noise  + (size_t)row * NUM_CLASSES;
    if (t < NUM_CLASSES / 4) {                       // 250 x 16B per array
        uint64_t ga = (uint64_t)(uintptr_t)(gl + t * 4);
        uint32_t la = (uint32_t)(uintptr_t)(&sLog[t * 4]);
        asm volatile("global_load_async_to_lds_b128 %0, %1, off"
                     :: "v"(la), "v"(ga) : "memory");
        uint64_t gb = (uint64_t)(uintptr_t)(gn + t * 4);
        uint32_t lb = (uint32_t)(uintptr_t)(&sNoise[t * 4]);
        asm volatile("global_load_async_to_lds_b128 %0, %1, off"
                     :: "v"(lb), "v"(gb) : "memory");
    }
    asm volatile("s_wait_asynccnt 0" ::: "memory");
    __syncthreads();

    // ---- Load my 4 contiguous elements into registers ----
    float4 lg4 = reinterpret_cast<const float4*>(sLog)[t];
    float4 nz4 = reinterpret_cast<const float4*>(sNoise)[t];
    float raw[4], nz[4];
    bool  val[4];
    {
        float lv[4] = {lg4.x, lg4.y, lg4.z, lg4.w};
        float nv[4] = {nz4.x, nz4.y, nz4.z, nz4.w};
#pragma unroll
        for (int k = 0; k < 4; ++k) {
            val[k] = (j0 + k) < NUM_CLASSES;
            raw[k] = val[k] ? (lv[k] / 6.0f) : NEGBIG;
            nz[k]  = val[k] ? nv[k] : 0.0f;
        }
    }

    // ---- softmax1: only max_conf = 1 / sum(exp(raw - max)) is needed ----
    float m1 = NEGBIG;
#pragma unroll
    for (int k = 0; k < 4; ++k) m1 = fmaxf(m1, raw[k]);
    const float M1 = blockMax(m1, sRed);
    float s1 = 0.0f;
#pragma unroll
    for (int k = 0; k < 4; ++k) s1 += val[k] ? __expf(raw[k] - M1) : 0.0f;
    const float S1 = blockSum(s1, sRed);
    const float max_conf = 1.0f / S1;
    const float stddev = 0.3f + 0.6f * max_conf * max_conf;

    // ---- noisy logits, softmax2 ----
    float nsy[4];
    float m2 = NEGBIG;
#pragma unroll
    for (int k = 0; k < 4; ++k) {
        nsy[k] = val[k] ? (raw[k] + nz[k] * stddev) : NEGBIG;
        m2 = fmaxf(m2, nsy[k]);
    }
    const float M2 = blockMax(m2, sRed);
    float e[4];
    float s2 = 0.0f;
#pragma unroll
    for (int k = 0; k < 4; ++k) {
        e[k] = val[k] ? __expf(nsy[k] - M2) : 0.0f;
        s2 += e[k];
    }
    const float S2 = blockSum(s2, sRed);
    const float invS2 = 1.0f / S2;

    // ---- clip, renormalize, round to tenths ----
    float c[4], cl = 0.0f;
#pragma unroll
    for (int k = 0; k < 4; ++k) {
        c[k] = fminf(e[k] * invS2, 0.6f);
        cl += c[k];
    }
    const float T = blockSum(cl, sRed);
    const float invT = 1.0f / T;
    float r[4], ls = 0.0f;
#pragma unroll
    for (int k = 0; k < 4; ++k) {
        r[k] = __builtin_rintf(c[k] * invT * 10.0f) * 0.1f;   // round-half-even
        ls += r[k];
    }
    const float R = blockSum(ls, sRed);
    if (R == 0.0f) {                       // all-zero row -> uniform
        ls = 0.0f;
#pragma unroll
        for (int k = 0; k < 4; ++k) {
            r[k] = val[k] ? (1.0f / NUM_CLASSES) : 0.0f;
            ls += r[k];
        }
    }

    // ---- 256-wide inclusive scan of thread partials via V_WMMA_F32_16X16X4_F32
    // X[16][16] row-major = partials; Y = X*U + Lstrict*(rowsum broadcast).
    sPart[t] = ls;
    __syncthreads();
    if ((t >> 5) == 0) {                   // wave 0, all 32 lanes active (EXEC=~0)
        const int L    = t;
        const int half = L >> 4;           // A/B K-halves live in lane halves
        const int Lm   = L & 15;
        v8f acc = {0.f, 0.f, 0.f, 0.f, 0.f, 0.f, 0.f, 0.f};
        // acc = X * U   (U upper-triangular ones -> within-row inclusive scan)
#pragma unroll
        for (int s = 0; s < 4; ++s) {
            const int k0 = 4 * s + 2 * half;
            v2f a, b;
            a.x = sPart[16 * Lm + k0];
            a.y = sPart[16 * Lm + k0 + 1];
            b.x = (k0     <= Lm) ? 1.0f : 0.0f;
            b.y = (k0 + 1 <= Lm) ? 1.0f : 0.0f;
            acc = __builtin_amdgcn_wmma_f32_16x16x4_f32(
                false, a, false, b, (short)0, acc, false, false);
        }
        // stash row sums (column 15 of acc) for the carry matmul
        if (Lm == 15) {
#pragma unroll
            for (int v = 0; v < 8; ++v) sRow[v + 8 * half] = acc[v];
        }
        // acc += Lstrict * (rowsum replicated across columns)  -> carry-in
#pragma unroll
        for (int s = 0; s < 4; ++s) {
            const int k0 = 4 * s + 2 * half;
            v2f a, b;
            a.x = (k0     < Lm) ? 1.0f : 0.0f;
            a.y = (k0 + 1 < Lm) ? 1.0f : 0.0f;
            b.x = sRow[k0];
            b.y = sRow[k0 + 1];
            acc = __builtin_amdgcn_wmma_f32_16x16x4_f32(
                false, a, false, b, (short)0, acc, false, false);
        }
#pragma unroll
        for (int v = 0; v < 8; ++v) sScan[16 * (v + 8 * half) + Lm] = acc[v];
    }
    __syncthreads();

    // ---- inverse-CDF sample: idx = count(cum < u * total) ----
    const float total  = sScan[BLOCK - 1];
    const float thresh = u[row] * total;
    float cum = sScan[t] - ls;             // exclusive offset for this thread
    int cnt = 0;
#pragma unroll
    for (int k = 0; k < 4; ++k) {
        cum += r[k];
        cnt += (cum < thresh) ? 1 : 0;
    }
    const float idxf = blockSum((float)cnt, sRed);
    const int idx = (int)(idxf + 0.5f);

    // ---- write log-one-hot row (float4 stores, 16B aligned) ----
    if (t < NUM_CLASSES / 4) {
        float o[4];
#pragma unroll
        for (int k = 0; k < 4; ++k) o[k] = ((j0 + k) == idx) ? OUT_HI : OUT_LO;
        float4 o4 = make_float4(o[0], o[1], o[2], o[3]);
        reinterpret_cast<float4*>(out + (size_t)row * NUM_CLASSES)[t] = o4;
    }
}

extern "C" void kernel_launch(void* const* d_in, const int* in_sizes, int n_in,
                              void* d_out, int out_size, void* d_ws, size_t ws_size,
                              hipStream_t stream) {
    const float* logits = (const float*)d_in[0];
    const float* noise  = (const float*)d_in[1];
    const float* u      = (const float*)d_in[2];
    float* out = (float*)d_out;
    const int rows = in_sizes[2];          // u has one entry per row
    sanitize_sample_kernel<<<dim3(rows), dim3(BLOCK), 0, stream>>>(logits, noise, u, out);
}